// RRN_70300024701635
// MI455X (gfx1250) — compile-verified
//
#include <hip/hip_runtime.h>
#include <math.h>

// Problem constants (match reference)
#define M_ROWS 131072
#define KCLS   64
#define DDIM   128
#define LLAY   8
#define TLTRI  4096

typedef float v2f __attribute__((ext_vector_type(2)));
typedef float v8f __attribute__((ext_vector_type(8)));

// D = A(16x4 f32) * B(4x16 f32) + C(16x16 f32), wave32 CDNA5 WMMA.
__device__ __forceinline__ v8f wmma_f32(v2f a, v2f b, v8f c) {
  return __builtin_amdgcn_wmma_f32_16x16x4_f32(
      /*neg_a=*/false, a, /*neg_b=*/false, b,
      /*c_mod=*/(short)0, c, /*reuse_a=*/false, /*reuse_b=*/false);
}

__device__ __forceinline__ float sigmoidf_fast(float x) {
  return 1.0f / (1.0f + __expf(-x));
}

// Sum across the 16 lanes of each wave-half; every lane of the half gets the sum.
__device__ __forceinline__ float half16_reduce(float p) {
  p += __shfl_xor(p, 8, 32);
  p += __shfl_xor(p, 4, 32);
  p += __shfl_xor(p, 2, 32);
  p += __shfl_xor(p, 1, 32);
  return p;
}

// ---------------------------------------------------------------------------
// ClassUpdate: E <- l2norm(E + sigmoid(cat@V^T) * relu(cat@W^T)), cat=[E|memb]
// One block = 16 rows. 8 waves each compute a 16x16 tile of both GEMM outputs
// over K=192 with fp32 WMMA, then cooperative row-wise L2 norm in LDS.
// ---------------------------------------------------------------------------
__global__ __launch_bounds__(256)
void rrn_class_update(const float* __restrict__ Esrc,
                      const float* __restrict__ memb,
                      const float* __restrict__ clsV,   // [128,192] row-major
                      const float* __restrict__ clsW,   // [128,192] row-major
                      float* __restrict__ Edst) {
  __shared__ float cat[16][200];   // 192 + pad
  __shared__ float sout[16][136];  // 128 + pad

  const int tid  = threadIdx.x;
  const int row0 = blockIdx.x * 16;

  // Stage cat = [E | memberships] for 16 rows.
  for (int idx = tid; idx < 16 * 192; idx += 256) {
    const int r = idx / 192, c = idx % 192;
    cat[r][c] = (c < DDIM)
        ? Esrc[(size_t)(row0 + r) * DDIM + c]
        : memb[(size_t)(row0 + r) * KCLS + (c - DDIM)];
  }
  __syncthreads();

  const int lane = tid & 31;
  const int wave = tid >> 5;
  const int half = lane >> 4;   // 0: lanes 0-15, 1: lanes 16-31
  const int mn   = lane & 15;   // row (A) / col (B) index within tile
  const int n0   = wave * 16;   // this wave's output-column tile

  v8f accV = {};
  v8f accW = {};
  for (int kk = 0; kk < 192; kk += 4) {
    const int ko = kk + 2 * half;
    const v2f a  = *(const v2f*)&cat[mn][ko];
    const v2f bV = *(const v2f*)(clsV + (size_t)(n0 + mn) * 192 + ko);
    const v2f bW = *(const v2f*)(clsW + (size_t)(n0 + mn) * 192 + ko);
    accV = wmma_f32(a, bV, accV);
    accW = wmma_f32(a, bW, accW);
  }

  // Gated residual into LDS.
#pragma unroll
  for (int r = 0; r < 8; ++r) {
    const int m = r + 8 * half;
    const int n = n0 + mn;
    const float g = sigmoidf_fast(accV[r]);
    const float d = fmaxf(accW[r], 0.0f);
    sout[m][n] = cat[m][n] + g * d;
  }
  __syncthreads();

  // Row-wise L2 normalize, 16 threads per row.
  const int rr = tid >> 4;
  const int j  = tid & 15;
  float p = 0.0f;
  for (int c = j; c < DDIM; c += 16) { const float v = sout[rr][c]; p += v * v; }
  p = half16_reduce(p);
  const float scale = 1.0f / fmaxf(sqrtf(p), 1e-12f);
  for (int c = j; c < DDIM; c += 16)
    Edst[(size_t)(row0 + rr) * DDIM + c] = sout[rr][c] * scale;
}

// ---------------------------------------------------------------------------
// RelationUpdate: one block = 16 triples of one layer. Gather es/eo, run the
// 8 per-layer GEMMs as 4 fused WMMA accumulator chains, epilogue + L2 norm,
// scatter new_s/new_o back into E (indices globally unique -> in-place safe).
// ---------------------------------------------------------------------------
__global__ __launch_bounds__(256)
void rrn_relation_update(float* __restrict__ E,
                         const float* __restrict__ subVs, const float* __restrict__ subVo,
                         const float* __restrict__ subWs, const float* __restrict__ subWo,
                         const float* __restrict__ subw,
                         const float* __restrict__ objVs, const float* __restrict__ objVo,
                         const float* __restrict__ objWs, const float* __restrict__ objWo,
                         const float* __restrict__ objw,
                         const int* __restrict__ subjIdx, const int* __restrict__ objIdx) {
  __shared__ float es[16][136];
  __shared__ float eo[16][136];
  __shared__ float ns[16][136];
  __shared__ float no_[16][136];
  __shared__ float dotS[16];
  __shared__ float dotO[16];
  __shared__ int   sIdx[16];
  __shared__ int   oIdx[16];

  const int tid      = threadIdx.x;
  const int tilesPerL = TLTRI / 16;                 // 256
  const int l  = blockIdx.x / tilesPerL;
  const int t0 = (blockIdx.x % tilesPerL) * 16;

  const size_t wOff = (size_t)l * DDIM * DDIM;
  const float* Vs  = subVs + wOff;
  const float* Vo  = subVo + wOff;
  const float* Ws  = subWs + wOff;
  const float* Wo  = subWo + wOff;
  const float* oVs = objVs + wOff;
  const float* oVo = objVo + wOff;
  const float* oWs = objWs + wOff;
  const float* oWo = objWo + wOff;
  const float* sw  = subw + (size_t)l * DDIM;
  const float* ow  = objw + (size_t)l * DDIM;

  if (tid < 16)       sIdx[tid]      = subjIdx[(size_t)l * TLTRI + t0 + tid];
  else if (tid < 32)  oIdx[tid - 16] = objIdx[(size_t)l * TLTRI + t0 + (tid - 16)];
  __syncthreads();

  // Gather es/eo tiles (float4 vectorized; rows are 512B-aligned in global,
  // LDS row stride 136 floats = 544B is 16B-aligned).
  for (int idx = tid; idx < 16 * 32; idx += 256) {
    const int r = idx >> 5, c4 = idx & 31;
    ((float4*)&es[r][0])[c4] = ((const float4*)(E + (size_t)sIdx[r] * DDIM))[c4];
    ((float4*)&eo[r][0])[c4] = ((const float4*)(E + (size_t)oIdx[r] * DDIM))[c4];
  }
  __syncthreads();

  // Per-triple interaction dots: dotS[t]=<eo_t, sub_w>, dotO[t]=<es_t, obj_w>.
  {
    const int rr = tid >> 4, j = tid & 15;
    float pS = 0.0f, pO = 0.0f;
    for (int c = j; c < DDIM; c += 16) {
      pS += eo[rr][c] * sw[c];
      pO += es[rr][c] * ow[c];
    }
    pS = half16_reduce(pS);
    pO = half16_reduce(pO);
    if (j == 0) { dotS[rr] = pS; dotO[rr] = pO; }
  }
  __syncthreads();

  const int lane = tid & 31;
  const int wave = tid >> 5;
  const int half = lane >> 4;
  const int mn   = lane & 15;
  const int n0   = wave * 16;
  const size_t bBase = (size_t)(n0 + mn) * DDIM;

  v8f accGS = {};  // es@Vs^T + eo@Vo^T  -> subject gate
  v8f accDS = {};  // es@Ws^T + eo@Wo^T  -> subject direction
  v8f accGO = {};  // es@oVs^T + eo@oVo^T -> object gate
  v8f accDO = {};  // es@oWs^T + eo@oWo^T -> object direction
  for (int kk = 0; kk < DDIM; kk += 4) {
    const int ko = kk + 2 * half;
    const v2f aS = *(const v2f*)&es[mn][ko];
    const v2f aO = *(const v2f*)&eo[mn][ko];
    const v2f bVs  = *(const v2f*)(Vs  + bBase + ko);
    const v2f bVo  = *(const v2f*)(Vo  + bBase + ko);
    const v2f bWs  = *(const v2f*)(Ws  + bBase + ko);
    const v2f bWo  = *(const v2f*)(Wo  + bBase + ko);
    const v2f bOVs = *(const v2f*)(oVs + bBase + ko);
    const v2f bOVo = *(const v2f*)(oVo + bBase + ko);
    const v2f bOWs = *(const v2f*)(oWs + bBase + ko);
    const v2f bOWo = *(const v2f*)(oWo + bBase + ko);
    accGS = wmma_f32(aS, bVs,  accGS);
    accGS = wmma_f32(aO, bVo,  accGS);
    accDS = wmma_f32(aS, bWs,  accDS);
    accDS = wmma_f32(aO, bWo,  accDS);
    accGO = wmma_f32(aS, bOVs, accGO);
    accGO = wmma_f32(aO, bOVo, accGO);
    accDO = wmma_f32(aS, bOWs, accDO);
    accDO = wmma_f32(aO, bOWo, accDO);
  }

  // Epilogue: gate * relu(dir + interaction), residual into LDS.
#pragma unroll
  for (int r = 0; r < 8; ++r) {
    const int m = r + 8 * half;
    const int n = n0 + mn;
    const float esv = es[m][n];
    const float eov = eo[m][n];
    const float gs = sigmoidf_fast(accGS[r]);
    const float ds = fmaxf(accDS[r] + esv * dotS[m], 0.0f);
    ns[m][n] = esv + gs * ds;
    const float go = sigmoidf_fast(accGO[r]);
    const float dv = fmaxf(accDO[r] + eov * dotO[m], 0.0f);
    no_[m][n] = eov + go * dv;
  }
  __syncthreads();

  // Row L2 norms + scatter (rows owned exclusively by this block).
  const int rr = tid >> 4;
  const int j  = tid & 15;
  float pS = 0.0f, pO = 0.0f;
  for (int c = j; c < DDIM; c += 16) {
    const float a = ns[rr][c];  pS += a * a;
    const float b = no_[rr][c]; pO += b * b;
  }
  pS = half16_reduce(pS);
  pO = half16_reduce(pO);
  const float scS = 1.0f / fmaxf(sqrtf(pS), 1e-12f);
  const float scO = 1.0f / fmaxf(sqrtf(pO), 1e-12f);
  float* rowS = E + (size_t)sIdx[rr] * DDIM;
  float* rowO = E + (size_t)oIdx[rr] * DDIM;
  for (int c = j; c < DDIM; c += 16) {
    rowS[c] = ns[rr][c] * scS;
    rowO[c] = no_[rr][c] * scO;
  }
}

// ---------------------------------------------------------------------------
extern "C" void kernel_launch(void* const* d_in, const int* in_sizes, int n_in,
                              void* d_out, int out_size, void* d_ws, size_t ws_size,
                              hipStream_t stream) {
  (void)in_sizes; (void)n_in; (void)out_size; (void)d_ws; (void)ws_size;
  const float* embedding_m = (const float*)d_in[0];
  const float* memberships = (const float*)d_in[1];
  const float* cls_V = (const float*)d_in[2];
  const float* cls_W = (const float*)d_in[3];
  const float* sub_Vs = (const float*)d_in[4];
  const float* sub_Vo = (const float*)d_in[5];
  const float* sub_Ws = (const float*)d_in[6];
  const float* sub_Wo = (const float*)d_in[7];
  const float* sub_w  = (const float*)d_in[8];
  const float* obj_Vs = (const float*)d_in[9];
  const float* obj_Vo = (const float*)d_in[10];
  const float* obj_Ws = (const float*)d_in[11];
  const float* obj_Wo = (const float*)d_in[12];
  const float* obj_w  = (const float*)d_in[13];
  const int*   subj_idx = (const int*)d_in[14];
  const int*   obj_idx  = (const int*)d_in[15];
  float* E = (float*)d_out;

  const dim3 blk(256);
  const dim3 gridCls(M_ROWS / 16);            // 8192 blocks
  const dim3 gridRel((LLAY * TLTRI) / 16);    // 2048 blocks

  for (int iter = 0; iter < 2; ++iter) {
    const float* src = (iter == 0) ? embedding_m : E;
    rrn_class_update<<<gridCls, blk, 0, stream>>>(src, memberships, cls_V, cls_W, E);
    rrn_relation_update<<<gridRel, blk, 0, stream>>>(
        E, sub_Vs, sub_Vo, sub_Ws, sub_Wo, sub_w,
        obj_Vs, obj_Vo, obj_Ws, obj_Wo, obj_w, subj_idx, obj_idx);
  }
}